// Decoder1_73993696575813
// MI455X (gfx1250) — compile-verified
//
#include <hip/hip_runtime.h>
#include <hip/hip_bf16.h>
#include <math.h>

// Problem constants (from reference setup_inputs)
constexpr int kB  = 64;    // batch
constexpr int kT  = 512;   // seq length
constexpr int kV  = 2048;  // vocab
constexpr int kE  = 256;   // symbol embedding dim (E_SYM)
constexpr int kEE = 512;   // encoder embedding dim (E_ENC)
constexpr int kH  = 1024;  // hidden

typedef float v2f __attribute__((ext_vector_type(2)));
typedef float v8f __attribute__((ext_vector_type(8)));

__device__ __forceinline__ v2f load2(const float* p) {
  float2 t = *(const float2*)p;
  v2f r; r.x = t.x; r.y = t.y;
  return r;
}

__device__ __forceinline__ v8f wmma4(v2f a, v2f b, v8f c) {
  return __builtin_amdgcn_wmma_f32_16x16x4_f32(
      /*neg_a=*/false, a, /*neg_b=*/false, b,
      /*c_mod=*/(short)0, c, /*reuse_a=*/false, /*reuse_b=*/false);
}

// Single-tile K-loop (bridge kernel).
template <int K>
__device__ __forceinline__ v8f gemm_frag(const float* __restrict__ xp,
                                         const float* __restrict__ wp,
                                         v8f acc) {
  __builtin_prefetch(xp, 0, 0);
  __builtin_prefetch(wp, 0, 0);
#pragma unroll 8
  for (int k = 0; k < K; k += 4)
    acc = wmma4(load2(xp + k), load2(wp + k), acc);
  return acc;
}

// 4 M-tiles sharing one B (weight) fragment: 1.25 loads/WMMA, 4 independent
// accumulation chains for pipelining (GRU step).
template <int K>
__device__ __forceinline__ void gemm_frag4(
    const float* __restrict__ x0, const float* __restrict__ x1,
    const float* __restrict__ x2, const float* __restrict__ x3,
    const float* __restrict__ wp,
    v8f& a0, v8f& a1, v8f& a2, v8f& a3) {
  __builtin_prefetch(wp, 0, 0);
#pragma unroll 4
  for (int k = 0; k < K; k += 4) {
    const v2f b = load2(wp + k);
    a0 = wmma4(load2(x0 + k), b, a0);
    a1 = wmma4(load2(x1 + k), b, a1);
    a2 = wmma4(load2(x2 + k), b, a2);
    a3 = wmma4(load2(x3 + k), b, a3);
  }
}

// 2 M-tiles sharing one B fragment (logits kernel).
template <int K>
__device__ __forceinline__ void gemm_frag2(
    const float* __restrict__ x0, const float* __restrict__ x1,
    const float* __restrict__ wp, v8f& a0, v8f& a1) {
  __builtin_prefetch(wp, 0, 0);
#pragma unroll 4
  for (int k = 0; k < K; k += 4) {
    const v2f b = load2(wp + k);
    a0 = wmma4(load2(x0 + k), b, a0);
    a1 = wmma4(load2(x1 + k), b, a1);
  }
}

__device__ __forceinline__ float fsigmoid(float x) {
  return 1.0f / (1.0f + __expf(-x));
}

// -----------------------------------------------------------------------------
// Kernel 1: h0 = encoder_embedding @ bridge_W.T + bridge_b   [64,512]@[512,1024]
// 256 tiles of 16x16; 4 waves/block, one tile per wave. Tiny one-shot kernel.
// -----------------------------------------------------------------------------
__global__ __launch_bounds__(128) void bridge_kernel(
    const float* __restrict__ enc, const float* __restrict__ Wb,
    const float* __restrict__ bb, float* __restrict__ h0) {
  const int tid  = threadIdx.x;
  const int wave = tid >> 5;
  const int lane = tid & 31;
  const int idx  = lane & 15;
  const int half = lane >> 4;

  const int tile = blockIdx.x * 4 + wave;  // 0..255
  const int mt = tile & 3;                 // 4 row tiles  (B=64)
  const int nt = tile >> 2;                // 64 col tiles (H=1024)

  const float* xp = enc + (size_t)(mt * 16 + idx) * kEE + 2 * half;
  const float* wp = Wb  + (size_t)(nt * 16 + idx) * kEE + 2 * half;

  v8f acc = {};
  acc = gemm_frag<kEE>(xp, wp, acc);

  const float bias = bb[nt * 16 + idx];
#pragma unroll
  for (int i = 0; i < 8; ++i)
    h0[(size_t)(mt * 16 + i + 8 * half) * kH + nt * 16 + idx] = acc[i] + bias;
}

// -----------------------------------------------------------------------------
// Kernel 2: one GRU timestep. Block = hidden tile jt (64 blocks/step); each of
// the 4 waves owns one gate job across ALL 64 batch rows (4 M-tiles, shared
// weight fragment):
//   wave 0: r-sum = x@w_ih[r] + h@w_hh[r]   (fused accumulator)
//   wave 1: z-sum = x@w_ih[z] + h@w_hh[z]
//   wave 2: i_n   = x@w_ih[n]
//   wave 3: h_n   = h@w_hh[n]
// then in-block gate math -> h_out (ping-pong) and rnn_hidden output.
// -----------------------------------------------------------------------------
__global__ __launch_bounds__(128) void gru_step_kernel(
    int t, const float* __restrict__ h_in, float* __restrict__ h_out,
    float* __restrict__ hidden_out, const long long* __restrict__ y,
    const float* __restrict__ emb, const float* __restrict__ w_ih,
    const float* __restrict__ w_hh, const float* __restrict__ b_ih,
    const float* __restrict__ b_hh) {
  __shared__ float lds[16 * 256];  // [gate(4)][bi(4)][16x16]

  const int jt = blockIdx.x;       // 0..63
  const int j0 = jt * 16;
  const int tid  = threadIdx.x;
  const int wave = tid >> 5;
  const int lane = tid & 31;
  const int idx  = lane & 15;
  const int half = lane >> 4;
  const int koff = 2 * half;

  // Per-lane A-row pointers for the 4 batch tiles (token gather + h row).
  const float* xe[4];
  const float* xh[4];
#pragma unroll
  for (int bi = 0; bi < 4; ++bi) {
    const int arow = bi * 16 + idx;
    const long long tok = (t == 0) ? 0ll : y[(size_t)arow * kT + (t - 1)];
    xe[bi] = emb  + (size_t)tok * kE + koff;
    xh[bi] = h_in + (size_t)arow * kH + koff;
  }

  v8f a0 = {}, a1 = {}, a2 = {}, a3 = {};
  if (wave == 0) {
    const float* wi = w_ih + (size_t)(0 * kH + j0 + idx) * kE + koff;
    const float* wh = w_hh + (size_t)(0 * kH + j0 + idx) * kH + koff;
    gemm_frag4<kE>(xe[0], xe[1], xe[2], xe[3], wi, a0, a1, a2, a3);
    gemm_frag4<kH>(xh[0], xh[1], xh[2], xh[3], wh, a0, a1, a2, a3);
  } else if (wave == 1) {
    const float* wi = w_ih + (size_t)(1 * kH + j0 + idx) * kE + koff;
    const float* wh = w_hh + (size_t)(1 * kH + j0 + idx) * kH + koff;
    gemm_frag4<kE>(xe[0], xe[1], xe[2], xe[3], wi, a0, a1, a2, a3);
    gemm_frag4<kH>(xh[0], xh[1], xh[2], xh[3], wh, a0, a1, a2, a3);
  } else if (wave == 2) {
    const float* wi = w_ih + (size_t)(2 * kH + j0 + idx) * kE + koff;
    gemm_frag4<kE>(xe[0], xe[1], xe[2], xe[3], wi, a0, a1, a2, a3);
  } else {
    const float* wh = w_hh + (size_t)(2 * kH + j0 + idx) * kH + koff;
    gemm_frag4<kH>(xh[0], xh[1], xh[2], xh[3], wh, a0, a1, a2, a3);
  }

  {
    float* g = lds + wave * 4 * 256;
#pragma unroll
    for (int i = 0; i < 8; ++i) {
      const int e = (i + 8 * half) * 16 + idx;
      g[0 * 256 + e] = a0[i];
      g[1 * 256 + e] = a1[i];
      g[2 * 256 + e] = a2[i];
      g[3 * 256 + e] = a3[i];
    }
  }
  __syncthreads();

  // gate math: 64 x 16 = 1024 elements, 128 threads -> 8 each
  for (int e = tid; e < 1024; e += 128) {
    const int b  = e >> 4;         // 0..63
    const int n  = e & 15;
    const int bi = b >> 4;
    const int te = (b & 15) * 16 + n;
    const int j  = j0 + n;

    const float rs  = lds[(0 * 4 + bi) * 256 + te] + b_ih[j] + b_hh[j];
    const float zs  = lds[(1 * 4 + bi) * 256 + te] + b_ih[kH + j] + b_hh[kH + j];
    const float inn = lds[(2 * 4 + bi) * 256 + te] + b_ih[2 * kH + j];
    const float hnn = lds[(3 * 4 + bi) * 256 + te] + b_hh[2 * kH + j];

    const float r  = fsigmoid(rs);
    const float z  = fsigmoid(zs);
    const float nv = tanhf(inn + r * hnn);
    const float hp = h_in[(size_t)b * kH + j];
    const float hn = (1.0f - z) * nv + z * hp;

    h_out[(size_t)b * kH + j] = hn;
    hidden_out[((size_t)b * kT + t) * kH + j] = hn;
  }
}

// -----------------------------------------------------------------------------
// Kernel 3: fused logits + log_softmax. Block owns 32 rows of [B*T, V]:
// 8 waves WMMA-tile all V=2048 columns; each wave computes 2 M-tiles per
// fc_W fragment (halves weight re-read traffic). 32x2048 f32 tile lives in
// 256 KB dynamic LDS (WGP LDS = 320 KB); per-row log-softmax with wave
// shuffles. Logits never hit HBM.
// -----------------------------------------------------------------------------
__global__ __launch_bounds__(256) void logits_kernel(
    const float* __restrict__ hid, const float* __restrict__ fc,
    float* __restrict__ outp) {
  extern __shared__ float smem[];  // 32 * kV floats

  const int tid  = threadIdx.x;
  const int wave = tid >> 5;
  const int lane = tid & 31;
  const int idx  = lane & 15;
  const int half = lane >> 4;

  const int r0 = blockIdx.x * 32;  // first of 32 rows (row = b*T + t)
  const float* x0 = hid + (size_t)(r0 + idx) * kH + 2 * half;
  const float* x1 = hid + (size_t)(r0 + 16 + idx) * kH + 2 * half;

  for (int nt = wave; nt < kV / 16; nt += 8) {
    const float* wp = fc + (size_t)(nt * 16 + idx) * kH + 2 * half;
    v8f a0 = {}, a1 = {};
    gemm_frag2<kH>(x0, x1, wp, a0, a1);
#pragma unroll
    for (int i = 0; i < 8; ++i) {
      smem[(size_t)(i + 8 * half) * kV + nt * 16 + idx]      = a0[i];
      smem[(size_t)(16 + i + 8 * half) * kV + nt * 16 + idx] = a1[i];
    }
  }
  __syncthreads();

  // log-softmax: 32 rows, 8 waves -> 4 rows per wave
#pragma unroll
  for (int rep = 0; rep < 4; ++rep) {
    const int m = wave * 4 + rep;
    const float* row = smem + (size_t)m * kV;

    float mx = -INFINITY;
    for (int v = lane; v < kV; v += 32) mx = fmaxf(mx, row[v]);
#pragma unroll
    for (int off = 16; off; off >>= 1) mx = fmaxf(mx, __shfl_xor(mx, off, 32));

    float sum = 0.0f;
    for (int v = lane; v < kV; v += 32) sum += __expf(row[v] - mx);
#pragma unroll
    for (int off = 16; off; off >>= 1) sum += __shfl_xor(sum, off, 32);

    const float lse = mx + __logf(sum);
    float* orow = outp + (size_t)(r0 + m) * kV;
    for (int v = lane; v < kV; v += 32) orow[v] = row[v] - lse;
  }
}

// -----------------------------------------------------------------------------
extern "C" void kernel_launch(void* const* d_in, const int* in_sizes, int n_in,
                              void* d_out, int out_size, void* d_ws, size_t ws_size,
                              hipStream_t stream) {
  const float*     enc      = (const float*)d_in[0];
  const long long* y        = (const long long*)d_in[1];
  // d_in[2] = max_length (scalar, == kT, hardcoded)
  const float*     emb      = (const float*)d_in[3];
  const float*     bridge_W = (const float*)d_in[4];
  const float*     bridge_b = (const float*)d_in[5];
  const float*     w_ih     = (const float*)d_in[6];
  const float*     w_hh     = (const float*)d_in[7];
  const float*     b_ih     = (const float*)d_in[8];
  const float*     b_hh     = (const float*)d_in[9];
  const float*     fc_W     = (const float*)d_in[10];

  float* out_hidden = (float*)d_out;                              // [B,T,H]
  float* out_logp   = out_hidden + (size_t)kB * kT * kH;          // [B,T,V]

  // ping-pong hidden state in workspace (2 * 64*1024 f32 = 512 KB)
  float* h_ping = (float*)d_ws;
  float* h_pong = h_ping + (size_t)kB * kH;

  // 1) h0 = enc @ bridge_W.T + b
  bridge_kernel<<<64, 128, 0, stream>>>(enc, bridge_W, bridge_b, h_ping);

  // 2) sequential GRU: one kernel per timestep (graph-captured -> cheap replay)
  float* hin = h_ping;
  float* hout = h_pong;
  for (int t = 0; t < kT; ++t) {
    gru_step_kernel<<<64, 128, 0, stream>>>(
        t, hin, hout, out_hidden, y, emb, w_ih, w_hh, b_ih, b_hh);
    float* tmp = hin; hin = hout; hout = tmp;
  }

  // 3) logits GEMM + log_softmax fused (32 rows per block, 256 KB dynamic LDS)
  logits_kernel<<<kB * kT / 32, 256, 32 * kV * sizeof(float), stream>>>(
      out_hidden, fc_W, out_logp);
}